// Norm8PointNet_64175401337596
// MI455X (gfx1250) — compile-verified
//
#include <hip/hip_runtime.h>
#include <math.h>

#define B_     128
#define N_     8192
#define TOPK_  20
#define NS_    100
#define SQRT2_ 1.4142f

typedef float v2f __attribute__((ext_vector_type(2)));
typedef float v8f __attribute__((ext_vector_type(8)));

// ---------------------------------------------------------------------------
// Kernel 1: per-batch normalization stats (centroid + mean radial distance)
// par[b*8] = { s1, c1x, c1y, s2, c2x, c2y, 0, 0 }
// ---------------------------------------------------------------------------
__global__ void k_norm_stats(const float* __restrict__ p1,
                             const float* __restrict__ p2,
                             float* __restrict__ par) {
  const int b = blockIdx.x, tid = threadIdx.x;
  __shared__ float red[256];
  const float* P1 = p1 + (size_t)b * N_ * 3;
  const float* P2 = p2 + (size_t)b * N_ * 3;

  auto block_sum = [&](float v) -> float {
    red[tid] = v; __syncthreads();
    for (int s = 128; s > 0; s >>= 1) {
      if (tid < s) red[tid] += red[tid + s];
      __syncthreads();
    }
    float r = red[0]; __syncthreads();
    return r;
  };

  float sx1 = 0.f, sy1 = 0.f, sx2 = 0.f, sy2 = 0.f;
  for (int n = tid; n < N_; n += 256) {
    sx1 += P1[n * 3 + 0]; sy1 += P1[n * 3 + 1];
    sx2 += P2[n * 3 + 0]; sy2 += P2[n * 3 + 1];
  }
  const float cx1 = block_sum(sx1) / (float)N_;
  const float cy1 = block_sum(sy1) / (float)N_;
  const float cx2 = block_sum(sx2) / (float)N_;
  const float cy2 = block_sum(sy2) / (float)N_;

  float sd1 = 0.f, sd2 = 0.f;
  for (int n = tid; n < N_; n += 256) {
    float dx1 = P1[n * 3 + 0] - cx1, dy1 = P1[n * 3 + 1] - cy1;
    float dx2 = P2[n * 3 + 0] - cx2, dy2 = P2[n * 3 + 1] - cy2;
    sd1 += sqrtf(dx1 * dx1 + dy1 * dy1);
    sd2 += sqrtf(dx2 * dx2 + dy2 * dy2);
  }
  const float md1 = block_sum(sd1) / (float)N_;
  const float md2 = block_sum(sd2) / (float)N_;

  if (tid == 0) {
    float* q = par + b * 8;
    q[0] = SQRT2_ / md1; q[1] = cx1; q[2] = cy1;
    q[3] = SQRT2_ / md2; q[4] = cx2; q[5] = cy2;
    q[6] = 0.f; q[7] = 0.f;
  }
}

// ---------------------------------------------------------------------------
// Kernel 2: per-batch 9x9 Gram G = X^T X via V_WMMA_F32_16X16X4_F32.
// 256 threads / 8 waves per batch. A tile of 256 constraint rows (padded to
// 16 cols) is staged in LDS; each wave consumes its 32-row slice as eight
// K=4 WMMA steps, accumulating a 16x16 f32 tile in registers.
// A-operand layout (ISA 7.12.2, 32-bit A 16x4): lanes 0-15 hold K=0(.x),
// K=1(.y); lanes 16-31 hold K=2(.x), K=3(.y), col = lane%16. The B 4x16
// operand mirrors this transposed; since both operands are the same X chunk
// (G is symmetric), one fetch serves both.
// ---------------------------------------------------------------------------
__global__ void k_gram(const float* __restrict__ p1,
                       const float* __restrict__ p2,
                       const float* __restrict__ wv,
                       const float* __restrict__ par,
                       float* __restrict__ Gout) {
  const int b = blockIdx.x, tid = threadIdx.x;
  const int wave = tid >> 5, lane = tid & 31, half = lane >> 4, col = lane & 15;
  __shared__ float tile[256][16];
  __shared__ float Gpart[8][256];

  const float s1 = par[b * 8 + 0], cx1 = par[b * 8 + 1], cy1 = par[b * 8 + 2];
  const float s2 = par[b * 8 + 3], cx2 = par[b * 8 + 4], cy2 = par[b * 8 + 5];
  const float* P1 = p1 + (size_t)b * N_ * 3;
  const float* P2 = p2 + (size_t)b * N_ * 3;
  const float* W  = wv + (size_t)b * N_;

  v8f acc = {};
  for (int t = 0; t < N_ / 256; ++t) {
    const int n = t * 256 + tid;
    const float x1 = P1[n * 3 + 0], y1 = P1[n * 3 + 1], z1 = P1[n * 3 + 2];
    const float x2 = P2[n * 3 + 0], y2 = P2[n * 3 + 1], z2 = P2[n * 3 + 2];
    const float w = W[n];
    if (t + 1 < N_ / 256) {                      // next-tile prefetch (global_prefetch_b8)
      __builtin_prefetch(&P1[(n + 256) * 3], 0, 1);
      __builtin_prefetch(&P2[(n + 256) * 3], 0, 1);
    }
    const float a1 = s1 * (x1 - cx1 * z1), b1 = s1 * (y1 - cy1 * z1), c1 = z1;
    const float a2 = s2 * (x2 - cx2 * z2), b2 = s2 * (y2 - cy2 * z2);
    float r[9] = { a2 * a1, a2 * b1, a2 * c1,
                   b2 * a1, b2 * b1, b2 * c1,
                   a1,      b1,      c1 };
    float nn = 0.f;
    for (int i = 0; i < 9; ++i) nn += r[i] * r[i];
    const float scale = rsqrtf(nn) * w;
    for (int i = 0; i < 9; ++i) tile[tid][i] = r[i] * scale;
    for (int i = 9; i < 16; ++i) tile[tid][i] = 0.f;
    __syncthreads();

    for (int c = 0; c < 8; ++c) {
      const int k0 = wave * 32 + c * 4 + half * 2;
      v2f a;
      a.x = tile[k0 + 0][col];
      a.y = tile[k0 + 1][col];
      acc = __builtin_amdgcn_wmma_f32_16x16x4_f32(
          /*neg_a=*/false, a, /*neg_b=*/false, a,
          /*c_mod=*/(short)0, acc, /*reuse_a=*/false, /*reuse_b=*/false);
    }
    __syncthreads();
  }

  for (int r = 0; r < 8; ++r) Gpart[wave][(r + 8 * half) * 16 + col] = acc[r];
  __syncthreads();
  float g = 0.f;
  for (int w8 = 0; w8 < 8; ++w8) g += Gpart[w8][tid];
  Gout[(size_t)b * 256 + tid] = g;
}

// ---------------------------------------------------------------------------
// Scalar cyclic Jacobi eigensolver (symmetric), eigenvectors in columns of V.
// ---------------------------------------------------------------------------
__device__ void jacobi_eig(float* A, float* V, int n, int sweeps) {
  for (int i = 0; i < n * n; ++i) V[i] = 0.f;
  for (int i = 0; i < n; ++i) V[i * n + i] = 1.f;
  for (int sw = 0; sw < sweeps; ++sw) {
    for (int p = 0; p < n - 1; ++p) {
      for (int q = p + 1; q < n; ++q) {
        const float apq = A[p * n + q];
        if (fabsf(apq) < 1e-30f) continue;
        const float app = A[p * n + p], aqq = A[q * n + q];
        const float tau = (aqq - app) / (2.f * apq);
        const float t = (tau >= 0.f ? 1.f : -1.f) /
                        (fabsf(tau) + sqrtf(1.f + tau * tau));
        const float c = rsqrtf(1.f + t * t);
        const float s = t * c;
        for (int k = 0; k < n; ++k) {
          const float akp = A[k * n + p], akq = A[k * n + q];
          A[k * n + p] = c * akp - s * akq;
          A[k * n + q] = s * akp + c * akq;
        }
        for (int k = 0; k < n; ++k) {
          const float apk = A[p * n + k], aqk = A[q * n + k];
          A[p * n + k] = c * apk - s * aqk;
          A[q * n + k] = s * apk + c * aqk;
        }
        for (int k = 0; k < n; ++k) {
          const float vkp = V[k * n + p], vkq = V[k * n + q];
          V[k * n + p] = c * vkp - s * vkq;
          V[k * n + q] = s * vkp + c * vkq;
        }
      }
    }
  }
}

// ---------------------------------------------------------------------------
// Gram -> smallest eigenvector -> sign fix -> rank-2 projection -> denorm.
// F_ = F - (F v2) v2^T  ==  U diag(s1,s2,0) V^T. out = T2^T F_ T1.
// ---------------------------------------------------------------------------
__device__ void finish_solve(float* G, float s1, float cx1, float cy1,
                             float s2, float cx2, float cy2,
                             float* outF, float* vout) {
  float V[81];
  jacobi_eig(G, V, 9, 10);
  int jm = 0; float ev = G[0];
  for (int j = 1; j < 9; ++j) {
    const float e = G[j * 9 + j];
    if (e < ev) { ev = e; jm = j; }
  }
  float v[9];
  for (int i = 0; i < 9; ++i) v[i] = V[i * 9 + jm];
  int im = 0; float am = fabsf(v[0]);
  for (int i = 1; i < 9; ++i) {
    const float a = fabsf(v[i]);
    if (a > am) { am = a; im = i; }
  }
  const float sg = (v[im] > 0.f) ? 1.f : ((v[im] < 0.f) ? -1.f : 0.f);
  for (int i = 0; i < 9; ++i) v[i] *= sg;
  if (vout) for (int i = 0; i < 9; ++i) vout[i] = v[i];

  float F[9];
  for (int i = 0; i < 9; ++i) F[i] = v[i];

  float M[9];
  for (int i = 0; i < 3; ++i)
    for (int j = 0; j < 3; ++j) {
      float a = 0.f;
      for (int k = 0; k < 3; ++k) a += F[k * 3 + i] * F[k * 3 + j];
      M[i * 3 + j] = a;
    }
  float VM[9];
  jacobi_eig(M, VM, 3, 12);
  int j2 = 0; float e2 = M[0];
  for (int j = 1; j < 3; ++j) {
    const float e = M[j * 3 + j];
    if (e < e2) { e2 = e; j2 = j; }
  }
  const float w2[3] = { VM[0 * 3 + j2], VM[1 * 3 + j2], VM[2 * 3 + j2] };
  float Fw[3];
  for (int i = 0; i < 3; ++i)
    Fw[i] = F[i * 3 + 0] * w2[0] + F[i * 3 + 1] * w2[1] + F[i * 3 + 2] * w2[2];
  float Fp[9];
  for (int i = 0; i < 3; ++i)
    for (int j = 0; j < 3; ++j)
      Fp[i * 3 + j] = F[i * 3 + j] - Fw[i] * w2[j];

  const float T1[9] = { s1, 0.f, -cx1 * s1, 0.f, s1, -cy1 * s1, 0.f, 0.f, 1.f };
  const float T2[9] = { s2, 0.f, -cx2 * s2, 0.f, s2, -cy2 * s2, 0.f, 0.f, 1.f };
  float Tm[9];
  for (int i = 0; i < 3; ++i)
    for (int j = 0; j < 3; ++j) {
      float a = 0.f;
      for (int k = 0; k < 3; ++k) a += T2[k * 3 + i] * Fp[k * 3 + j];  // T2^T * Fp
      Tm[i * 3 + j] = a;
    }
  for (int i = 0; i < 3; ++i)
    for (int j = 0; j < 3; ++j) {
      float a = 0.f;
      for (int k = 0; k < 3; ++k) a += Tm[i * 3 + k] * T1[k * 3 + j];
      outF[i * 3 + j] = a;
    }
}

// ---------------------------------------------------------------------------
// Kernel 3: full-problem solves, one thread per batch.
// ---------------------------------------------------------------------------
__global__ void k_solve_full(const float* __restrict__ par,
                             const float* __restrict__ Gin,
                             float* __restrict__ out,
                             float* __restrict__ vstore) {
  const int p = blockIdx.x * blockDim.x + threadIdx.x;
  if (p >= B_) return;
  float G[81];
  for (int i = 0; i < 9; ++i)
    for (int j = 0; j < 9; ++j)
      G[i * 9 + j] = Gin[(size_t)p * 256 + i * 16 + j];
  float F[9], v[9];
  finish_solve(G, par[p * 8 + 0], par[p * 8 + 1], par[p * 8 + 2],
                  par[p * 8 + 3], par[p * 8 + 4], par[p * 8 + 5], F, v);
  for (int i = 0; i < 9; ++i) out[p * 9 + i] = F[i];
  for (int i = 0; i < 9; ++i) vstore[p * 16 + i] = v[i];
}

// ---------------------------------------------------------------------------
// Kernel 4: residual = X . v for the full problem (rows recomputed on the fly)
// ---------------------------------------------------------------------------
__global__ void k_residual(const float* __restrict__ p1,
                           const float* __restrict__ p2,
                           const float* __restrict__ wv,
                           const float* __restrict__ par,
                           const float* __restrict__ vstore,
                           float* __restrict__ res) {
  const int b = blockIdx.x >> 5;
  const int n = (blockIdx.x & 31) * 256 + threadIdx.x;
  const float s1 = par[b * 8 + 0], cx1 = par[b * 8 + 1], cy1 = par[b * 8 + 2];
  const float s2 = par[b * 8 + 3], cx2 = par[b * 8 + 4], cy2 = par[b * 8 + 5];
  const size_t base = ((size_t)b * N_ + n) * 3;
  const float x1 = p1[base], y1 = p1[base + 1], z1 = p1[base + 2];
  const float x2 = p2[base], y2 = p2[base + 1], z2 = p2[base + 2];
  const float w = wv[(size_t)b * N_ + n];
  const float a1 = s1 * (x1 - cx1 * z1), b1 = s1 * (y1 - cy1 * z1), c1 = z1;
  const float a2 = s2 * (x2 - cx2 * z2), b2 = s2 * (y2 - cy2 * z2);
  float r[9] = { a2 * a1, a2 * b1, a2 * c1,
                 b2 * a1, b2 * b1, b2 * c1,
                 a1,      b1,      c1 };
  float nn = 0.f;
  for (int i = 0; i < 9; ++i) nn += r[i] * r[i];
  const float scale = rsqrtf(nn) * w;
  float acc = 0.f;
  for (int i = 0; i < 9; ++i) acc += r[i] * scale * vstore[b * 16 + i];
  res[(size_t)b * N_ + n] = acc;
}

// ---------------------------------------------------------------------------
// Kernel 5: masked top-20 of weights per batch (iterative block argmax,
// ties -> smaller index, matching lax.top_k).
// ---------------------------------------------------------------------------
__global__ void k_topk(const float* __restrict__ wv,
                       const int* __restrict__ un,
                       int* __restrict__ sel_out) {
  const int b = blockIdx.x, tid = threadIdx.x;
  __shared__ float rv[256];
  __shared__ int   ri[256];
  __shared__ int   sel[TOPK_];
  const int m = un[b];
  const float* W = wv + (size_t)b * N_;
  for (int t = 0; t < TOPK_; ++t) {
    float bv = -INFINITY; int bi = 0x7fffffff;
    for (int n = tid; n < N_; n += 256) {
      if (n >= m) continue;
      bool used = false;
      for (int u = 0; u < t; ++u)
        if (sel[u] == n) { used = true; break; }
      if (used) continue;
      const float w = W[n];
      if (w > bv || (w == bv && n < bi)) { bv = w; bi = n; }
    }
    rv[tid] = bv; ri[tid] = bi; __syncthreads();
    for (int s = 128; s > 0; s >>= 1) {
      if (tid < s) {
        if (rv[tid + s] > rv[tid] ||
            (rv[tid + s] == rv[tid] && ri[tid + s] < ri[tid])) {
          rv[tid] = rv[tid + s]; ri[tid] = ri[tid + s];
        }
      }
      __syncthreads();
    }
    if (tid == 0) sel[t] = ri[0];
    __syncthreads();
  }
  if (tid < TOPK_) sel_out[b * TOPK_ + tid] = sel[tid];
}

// ---------------------------------------------------------------------------
// 20-point weighted-SVD solver (per-thread, used by top-k and sampled paths)
// ---------------------------------------------------------------------------
__device__ void solve20(const float* P1, const float* P2, const float* W,
                        const int* idx, float* outF, float* resid) {
  float q1[TOPK_][3], q2[TOPK_][3], w[TOPK_];
  float cx1 = 0.f, cy1 = 0.f, cx2 = 0.f, cy2 = 0.f;
  for (int k = 0; k < TOPK_; ++k) {
    const int n = idx[k];
    q1[k][0] = P1[n * 3]; q1[k][1] = P1[n * 3 + 1]; q1[k][2] = P1[n * 3 + 2];
    q2[k][0] = P2[n * 3]; q2[k][1] = P2[n * 3 + 1]; q2[k][2] = P2[n * 3 + 2];
    w[k] = W[n];
    cx1 += q1[k][0]; cy1 += q1[k][1];
    cx2 += q2[k][0]; cy2 += q2[k][1];
  }
  cx1 /= (float)TOPK_; cy1 /= (float)TOPK_;
  cx2 /= (float)TOPK_; cy2 /= (float)TOPK_;
  float md1 = 0.f, md2 = 0.f;
  for (int k = 0; k < TOPK_; ++k) {
    const float dx1 = q1[k][0] - cx1, dy1 = q1[k][1] - cy1;
    const float dx2 = q2[k][0] - cx2, dy2 = q2[k][1] - cy2;
    md1 += sqrtf(dx1 * dx1 + dy1 * dy1);
    md2 += sqrtf(dx2 * dx2 + dy2 * dy2);
  }
  md1 /= (float)TOPK_; md2 /= (float)TOPK_;
  const float s1 = SQRT2_ / md1, s2 = SQRT2_ / md2;

  float X[TOPK_][9];
  float G[81];
  for (int i = 0; i < 81; ++i) G[i] = 0.f;
  for (int k = 0; k < TOPK_; ++k) {
    const float z1 = q1[k][2], z2 = q2[k][2];
    const float a1 = s1 * (q1[k][0] - cx1 * z1), b1 = s1 * (q1[k][1] - cy1 * z1), c1 = z1;
    const float a2 = s2 * (q2[k][0] - cx2 * z2), b2 = s2 * (q2[k][1] - cy2 * z2);
    float r[9] = { a2 * a1, a2 * b1, a2 * c1,
                   b2 * a1, b2 * b1, b2 * c1,
                   a1,      b1,      c1 };
    float nn = 0.f;
    for (int i = 0; i < 9; ++i) nn += r[i] * r[i];
    const float scale = rsqrtf(nn) * w[k];
    for (int i = 0; i < 9; ++i) X[k][i] = r[i] * scale;
    for (int i = 0; i < 9; ++i)
      for (int j = 0; j < 9; ++j)
        G[i * 9 + j] += X[k][i] * X[k][j];
  }
  float F[9], v[9];
  finish_solve(G, s1, cx1, cy1, s2, cx2, cy2, F, v);
  for (int i = 0; i < 9; ++i) outF[i] = F[i];
  if (resid) {
    for (int k = 0; k < TOPK_; ++k) {
      float a = 0.f;
      for (int i = 0; i < 9; ++i) a += X[k][i] * v[i];
      resid[k] = a;
    }
  }
}

// ---------------------------------------------------------------------------
// Kernel 6: 128 top-k solves + 12800 sampled solves, one thread each.
// ---------------------------------------------------------------------------
__global__ void k_small(const float* __restrict__ p1,
                        const float* __restrict__ p2,
                        const float* __restrict__ wv,
                        const int* __restrict__ topk_idx,
                        const int* __restrict__ sampled,
                        float* __restrict__ outk,
                        float* __restrict__ resk,
                        float* __restrict__ outs) {
  const int p = blockIdx.x * blockDim.x + threadIdx.x;
  const int total = B_ + B_ * NS_;
  if (p >= total) return;
  int b; const int* idx; float* oF; float* oR = nullptr;
  if (p < B_) {
    b = p;
    idx = topk_idx + b * TOPK_;
    oF = outk + b * 9;
    oR = resk + b * TOPK_;
  } else {
    const int q = p - B_;
    b = q / NS_;
    const int s = q % NS_;
    idx = sampled + (size_t)(b * NS_ + s) * TOPK_;
    oF = outs + (size_t)(b * NS_ + s) * 9;
  }
  solve20(p1 + (size_t)b * N_ * 3, p2 + (size_t)b * N_ * 3,
          wv + (size_t)b * N_, idx, oF, oR);
}

// ---------------------------------------------------------------------------
// Kernel 7: w_accu = normalize(prod_k(w*1000)) per batch.
// ---------------------------------------------------------------------------
__global__ void k_waccu(const float* __restrict__ wv,
                        const int* __restrict__ sampled,
                        float* __restrict__ out) {
  const int b = blockIdx.x, tid = threadIdx.x;
  __shared__ float pr[NS_];
  __shared__ float ssum;
  if (tid < NS_) {
    float p = 1.f;
    for (int k = 0; k < TOPK_; ++k)
      p *= wv[(size_t)b * N_ + sampled[(size_t)(b * NS_ + tid) * TOPK_ + k]] * 1000.f;
    pr[tid] = p;
  }
  __syncthreads();
  if (tid == 0) {
    float s = 0.f;
    for (int i = 0; i < NS_; ++i) s += pr[i];
    ssum = s + 1e-10f;
  }
  __syncthreads();
  if (tid < NS_) out[b * NS_ + tid] = pr[tid] / ssum;
}

// ---------------------------------------------------------------------------
extern "C" void kernel_launch(void* const* d_in, const int* in_sizes, int n_in,
                              void* d_out, int out_size, void* d_ws, size_t ws_size,
                              hipStream_t stream) {
  const float* pts1    = (const float*)d_in[0];
  const float* pts2    = (const float*)d_in[1];
  const float* weights = (const float*)d_in[2];   // (B,1,N) flat == (B,N)
  const int*   un      = (const int*)d_in[3];
  const int*   sampled = (const int*)d_in[4];

  float* O       = (float*)d_out;
  float* O_out   = O;                   // (B,3,3)      1152
  float* O_res   = O + 1152;            // (B,N)        1048576
  float* O_outk  = O + 1049728;         // (B,3,3)      1152
  float* O_resk  = O + 1050880;         // (B,20)       2560
  float* O_outs  = O + 1053440;         // (B,100,3,3)  115200
  float* O_waccu = O + 1168640;         // (B,100,1)    12800

  float* W   = (float*)d_ws;
  float* PAR = W;                       // B*8
  float* GM  = W + 1024;                // B*256
  float* VST = W + 33792;               // B*16
  int*   TI  = (int*)(W + 35840);       // B*20 ints

  k_norm_stats<<<B_, 256, 0, stream>>>(pts1, pts2, PAR);
  k_gram      <<<B_, 256, 0, stream>>>(pts1, pts2, weights, PAR, GM);
  k_topk      <<<B_, 256, 0, stream>>>(weights, un, TI);
  k_solve_full<<<4, 32, 0, stream>>>(PAR, GM, O_out, VST);
  k_residual  <<<B_ * 32, 256, 0, stream>>>(pts1, pts2, weights, PAR, VST, O_res);
  {
    const int total = B_ + B_ * NS_;   // 12928
    k_small<<<(total + 127) / 128, 128, 0, stream>>>(
        pts1, pts2, weights, TI, sampled, O_outk, O_resk, O_outs);
  }
  k_waccu<<<B_, 128, 0, stream>>>(weights, sampled, O_waccu);
}